// FraudGraphSAGE_28570122453616
// MI455X (gfx1250) — compile-verified
//
#include <hip/hip_runtime.h>
#include <hip/hip_bf16.h>
#include <hip/hip_fp16.h>

typedef __attribute__((ext_vector_type(2))) float v2f;
typedef __attribute__((ext_vector_type(8))) float v8f;

// ---------------------------------------------------------------------------
// Degree / normalization kernels
// ---------------------------------------------------------------------------
__global__ void k_init_deg(float* __restrict__ deg, int N) {
  int i = blockIdx.x * blockDim.x + threadIdx.x;
  if (i < N) deg[i] = 1.0f;  // self-loop contributes 1 to in-degree
}

__global__ void k_accum_deg(const int* __restrict__ dst, int E, float* __restrict__ deg) {
  int e = blockIdx.x * blockDim.x + threadIdx.x;
  if (e < E) atomicAdd(&deg[dst[e]], 1.0f);
}

__global__ void k_dinv(float* __restrict__ deg, int N) {
  int i = blockIdx.x * blockDim.x + threadIdx.x;
  if (i < N) deg[i] = rsqrtf(deg[i]);  // deg >= 1 always (self-loops)
}

// ---------------------------------------------------------------------------
// WMMA GEMM: one wave per 16-row M-tile, ALL N-tiles accumulated in registers
// (A operand read exactly once). W is staged once per block into LDS,
// transposed with +2 padding so a B fragment (two consecutive K for fixed n)
// is one aligned ds_load_b64.
//
//   out[m, n] = dinv[m] * sum_k A[m,k] * W[k,n]
// PRE=false: A[m,k] = X[m,k]                      (layer 1, X = node features)
// PRE=true : A[m,k] = relu(dinv[m]*(ACC[m,k]+X[m,k]) + bias[k])  (fused layer-1
//            epilogue feeding layer 2; each element is computed exactly once)
// ---------------------------------------------------------------------------
template <int K, int NOUT, bool PRE>
__global__ void k_gemm_wmma(const float* __restrict__ X, const float* __restrict__ ACC,
                            const float* __restrict__ bias, const float* __restrict__ W,
                            const float* __restrict__ dinv, float* __restrict__ out,
                            int Mtiles) {
  constexpr int Ntiles = NOUT / 16;
  constexpr int KP = K + 2;                 // padded LDS row (8B-aligned b64 frags)
  __shared__ float lw[NOUT * KP];

  // Stage W transposed: lw[n*KP + k] = W[k*NOUT + n]  (coalesced global reads)
  for (int idx = threadIdx.x; idx < K * NOUT; idx += blockDim.x) {
    const int n = idx % NOUT;
    const int k = idx / NOUT;
    lw[n * KP + k] = W[idx];
  }
  __syncthreads();

  const int lane = threadIdx.x & 31;
  const int wave = threadIdx.x >> 5;
  const int mt = blockIdx.x * (blockDim.x >> 5) + wave;
  if (mt >= Mtiles) return;
  const int half = lane >> 4;   // 0: lanes 0-15 (K=kb,kb+1), 1: lanes 16-31 (K=kb+2,kb+3)
  const int l16  = lane & 15;

  const int arowi = mt * 16 + l16;                   // A fragment row: M = l16
  const float* xr = X + (size_t)arowi * K;
  const float* ar = PRE ? (ACC + (size_t)arowi * K) : nullptr;
  const float  di = PRE ? dinv[arowi] : 0.0f;

  v8f c[Ntiles] = {};

  for (int kb = 0; kb < K; kb += 4) {
    const int ka = kb + 2 * half;
    v2f a = *(const v2f*)&xr[ka];
    if (PRE) {
      const v2f ac = *(const v2f*)&ar[ka];
      float v0 = di * (a.x + ac.x) + bias[ka];
      float v1 = di * (a.y + ac.y) + bias[ka + 1];
      a.x = v0 > 0.0f ? v0 : 0.0f;
      a.y = v1 > 0.0f ? v1 : 0.0f;
    }
#pragma unroll
    for (int t = 0; t < Ntiles; ++t) {
      const int n = t * 16 + l16;
      const v2f b = *(const v2f*)&lw[n * KP + ka];   // ds_load_b64
      c[t] = __builtin_amdgcn_wmma_f32_16x16x4_f32(
          /*neg_a=*/false, a, /*neg_b=*/false, b,
          /*c_mod=*/(short)0, c[t], /*reuse_a=*/false, /*reuse_b=*/false);
    }
  }

  // C/D layout: lane<16 -> M=v, N=lane ; lane>=16 -> M=v+8, N=lane-16
#pragma unroll
  for (int t = 0; t < Ntiles; ++t) {
#pragma unroll
    for (int v = 0; v < 8; ++v) {
      const int m = mt * 16 + v + 8 * half;
      out[(size_t)m * NOUT + t * 16 + l16] = c[t][v] * dinv[m];
    }
  }
}

// ---------------------------------------------------------------------------
// Edge scatter: acc[dst[e], :] += hs[src[e], :]   (float4 per thread;
// 153M f32 atomics total -> L2-resident, the workload's true bottleneck)
// ---------------------------------------------------------------------------
template <int C>
__global__ void k_scatter(const float* __restrict__ hs, const int* __restrict__ src,
                          const int* __restrict__ dst, float* __restrict__ acc, int E) {
  constexpr int C4 = C / 4;
  const long long idx = (long long)blockIdx.x * blockDim.x + threadIdx.x;
  const int e  = (int)(idx / C4);
  const int c4 = (int)(idx % C4);
  if (e >= E) return;
  const int s = src[e];
  const int d = dst[e];
  const float4 m = *(const float4*)(hs + (size_t)s * C + c4 * 4);
  float* ap = acc + (size_t)d * C + c4 * 4;
  atomicAdd(ap + 0, m.x);
  atomicAdd(ap + 1, m.y);
  atomicAdd(ap + 2, m.z);
  atomicAdd(ap + 3, m.w);
}

// ---------------------------------------------------------------------------
// Fused layer-2 epilogue + classifier head:
//   t[c]  = relu( dinv[i]*(acc[i,c] + hs[i,c]) + b2[c] )
//   out_i = sigmoid( t . Wc + bc )
// ---------------------------------------------------------------------------
__global__ void k_cls(const float* __restrict__ hs, const float* __restrict__ acc,
                      const float* __restrict__ dinv, const float* __restrict__ b2,
                      const float* __restrict__ Wc, const float* __restrict__ bc,
                      float* __restrict__ out, int N) {
  const int i = blockIdx.x * blockDim.x + threadIdx.x;
  if (i >= N) return;
  const float di = dinv[i];
  const float4* hp = (const float4*)(hs + (size_t)i * 32);
  const float4* ap = (const float4*)(acc + (size_t)i * 32);
  const float4* wp = (const float4*)Wc;
  const float4* bp = (const float4*)b2;
  float s = 0.0f;
#pragma unroll
  for (int j = 0; j < 8; ++j) {
    const float4 h = hp[j], a = ap[j], w = wp[j], b = bp[j];
    const float v0 = fmaxf(di * (a.x + h.x) + b.x, 0.0f);
    const float v1 = fmaxf(di * (a.y + h.y) + b.y, 0.0f);
    const float v2 = fmaxf(di * (a.z + h.z) + b.z, 0.0f);
    const float v3 = fmaxf(di * (a.w + h.w) + b.w, 0.0f);
    s += v0 * w.x + v1 * w.y + v2 * w.z + v3 * w.w;
  }
  s += bc[0];
  out[i] = 1.0f / (1.0f + __expf(-s));
}

// ---------------------------------------------------------------------------
extern "C" void kernel_launch(void* const* d_in, const int* in_sizes, int n_in,
                              void* d_out, int out_size, void* d_ws, size_t ws_size,
                              hipStream_t stream) {
  const float* x  = (const float*)d_in[0];
  const int*   ei = (const int*)d_in[1];
  // d_in[2] = num_nodes scalar (unused; derived from sizes)
  const float* W1 = (const float*)d_in[3];
  const float* b1 = (const float*)d_in[4];
  const float* W2 = (const float*)d_in[5];
  const float* b2 = (const float*)d_in[6];
  const float* Wc = (const float*)d_in[7];
  const float* bc = (const float*)d_in[8];

  const int N = in_sizes[0] / 128;
  const int E = in_sizes[1] / 2;
  const int* src = ei;
  const int* dst = ei + E;

  // Workspace layout (floats), 16B-aligned slices
  float* ws = (float*)d_ws;
  size_t off = 0;
  float* dinv = ws + off; off += ((size_t)N + 1023) & ~(size_t)1023;
  float* h1   = ws + off; off += (size_t)N * 64;   // dinv-scaled layer-1 GEMM rows
  float* acc1 = ws + off; off += (size_t)N * 64;
  float* h2   = ws + off; off += (size_t)N * 32;   // dinv-scaled layer-2 GEMM rows
  float* acc2 = ws + off; off += (size_t)N * 32;
  (void)ws_size; (void)n_in; (void)out_size;

  const int T = 256;            // 8 waves per block
  const int Mtiles = (N + 15) / 16;
  const int gemmBlocks = (Mtiles + 7) / 8;

  // ---- degree normalization (shared by both layers) ----
  k_init_deg<<<(N + T - 1) / T, T, 0, stream>>>(dinv, N);
  k_accum_deg<<<(E + T - 1) / T, T, 0, stream>>>(dst, E, dinv);
  k_dinv<<<(N + T - 1) / T, T, 0, stream>>>(dinv, N);

  // ---- layer 1: h1 = dinv .* (x @ W1) ----
  k_gemm_wmma<128, 64, false><<<gemmBlocks, T, 0, stream>>>(
      x, nullptr, nullptr, W1, dinv, h1, Mtiles);
  hipMemsetAsync(acc1, 0, (size_t)N * 64 * sizeof(float), stream);
  {
    const long long total = (long long)E * 16;  // E * C/4
    k_scatter<64><<<(unsigned)((total + T - 1) / T), T, 0, stream>>>(h1, src, dst, acc1, E);
  }

  // ---- layer 2 (layer-1 ReLU epilogue fused into the A-operand load):
  //      h2 = dinv .* ( relu(dinv.*(acc1 + h1) + b1) @ W2 ) ----
  k_gemm_wmma<64, 32, true><<<gemmBlocks, T, 0, stream>>>(
      h1, acc1, b1, W2, dinv, h2, Mtiles);
  hipMemsetAsync(acc2, 0, (size_t)N * 32 * sizeof(float), stream);
  {
    const long long total = (long long)E * 8;   // E * C/4
    k_scatter<32><<<(unsigned)((total + T - 1) / T), T, 0, stream>>>(h2, src, dst, acc2, E);
  }

  // ---- fused layer-2 epilogue + classifier head ----
  k_cls<<<(N + T - 1) / T, T, 0, stream>>>(h2, acc2, dinv, b2, Wc, bc, (float*)d_out, N);
}